// FrameExpander_70351564308608
// MI455X (gfx1250) — compile-verified
//
#include <hip/hip_runtime.h>
#include <hip/hip_bf16.h>

// ---------------------------------------------------------------------------
// FrameExpander on MI455X (gfx1250):
//   y = W(500x65536) @ x + b        -> 131 MB stream, HBM-bound (~5.6us @23.3TB/s)
//   z = y @ mask(kernels, kwidths)  -> 1 MB, parallelized over 16 WGs
// Stage 1: V_WMMA_F32_16X16X4_F32 as the dot-product engine (one WMMA per
// 16x4 f32 weight tile = 256 B of unique HBM traffic; matrix pipe is ~free at
// the 91 G WMMA/s needed to saturate HBM). Weight loads are non-temporal
// (streamed once); x is staged per-workgroup into LDS via async-to-LDS.
// ---------------------------------------------------------------------------

#define DIM_HID   65536
#define NKERS     500
#define ROWS_PAD  512                    // 500 padded to 32 tiles of 16
#define ROW_TILES (ROWS_PAD / 16)        // 32
#define KCHUNKS   32
#define CHUNK     (DIM_HID / KCHUNKS)    // 2048 columns per workgroup (8 KB)
#define WAVES     8
#define WAVE_K    (CHUNK / WAVES)        // 256 columns per wave
#define WAVE_IT   (WAVE_K / 4)           // 64 WMMAs per wave
#define ZSLICES   8
#define ZSLICE_W  63                     // ceil(500/8)

typedef __attribute__((ext_vector_type(2))) float v2f;
typedef __attribute__((ext_vector_type(4))) float v4f;
typedef __attribute__((ext_vector_type(8))) float v8f;
typedef __attribute__((ext_vector_type(4))) int   v4i;
// Address-space-qualified pointee types for the async-to-LDS builtin:
// diagnostic showed param0 = "int ext_vector(4) __device__(=AS1) *".
typedef __attribute__((address_space(1))) v4i gv4i;
typedef __attribute__((address_space(3))) v4i lv4i;

#if __has_builtin(__builtin_amdgcn_global_load_async_to_lds_b128) && \
    __has_builtin(__builtin_amdgcn_s_wait_asynccnt)
#define USE_ASYNC_LDS 1
#else
#define USE_ASYNC_LDS 0
#endif

// ---------------------------------------------------------------------------
// Stage 1: partial GEMV via WMMA f32 16x16x4.
// A layout (ISA 7.12.2, 32-bit A 16x4): lanes 0-15 hold M=0..15 with
// VGPR0=K0, VGPR1=K1; lanes 16-31 hold VGPR0=K2, VGPR1=K3.
// B is filled so B[k][n] == x[kbase+k] for every n: every C column then holds
// the identical row-dot-product, so lane 0 (rows m..m+7 in acc[0..7]) and
// lane 16 (rows m+8..m+15) carry the results regardless of B's exact N map.
// ---------------------------------------------------------------------------
__global__ void __launch_bounds__(256)
gemv_partials_wmma(const float* __restrict__ W, const float* __restrict__ x,
                   float* __restrict__ partials) {
    const int tile  = blockIdx.x;            // row tile 0..31
    const int chunk = blockIdx.y;            // K chunk  0..31
    const int lane  = threadIdx.x & 31;
    const int wave  = threadIdx.x >> 5;      // 0..7
    const int half  = lane >> 4;             // 0: K0/K1 lanes, 1: K2/K3 lanes
    const int mrow  = lane & 15;

    __shared__ float xs[CHUNK];              // staged input window (8 KB)
    __shared__ float red[WAVES * 16];

    // ---- stage x[chunk*CHUNK .. +CHUNK) into LDS --------------------------
    {
        const float* gx = x + (size_t)chunk * CHUNK;
        const int t4 = threadIdx.x * 4;      // 256 threads * 4 floats = 1024
#if USE_ASYNC_LDS
        __builtin_amdgcn_global_load_async_to_lds_b128(
            (gv4i*)(gx + t4),        (lv4i*)(xs + t4),        0, 0);
        __builtin_amdgcn_global_load_async_to_lds_b128(
            (gv4i*)(gx + 1024 + t4), (lv4i*)(xs + 1024 + t4), 0, 0);
        __builtin_amdgcn_s_wait_asynccnt(0);
#else
        *(v4f*)(xs + t4)        = *(const v4f*)(gx + t4);
        *(v4f*)(xs + 1024 + t4) = *(const v4f*)(gx + 1024 + t4);
#endif
    }
    __syncthreads();

    int row = tile * 16 + mrow;
    if (row >= NKERS) row = NKERS - 1;       // clamp (no divergence; WMMA needs EXEC all-1s)

    // Per-lane base pointers; loop is a compile-time-constant 64 trips so the
    // compiler can unroll and batch loads ahead of waits.
    const float* __restrict__ aw =
        W + (size_t)row * DIM_HID + chunk * CHUNK + wave * WAVE_K + half * 2;
    const float* __restrict__ xw = xs + wave * WAVE_K + half * 2;

    v8f acc = {};
    #pragma unroll 8
    for (int i = 0; i < WAVE_IT; ++i) {
        // A: float2 of this lane's row, streamed once -> non-temporal.
        v2f a = __builtin_nontemporal_load((const v2f*)(aw + i * 4));
        // B replicated across N: VGPR0 = x[K0]|x[K2] halves, VGPR1 = x[K1]|x[K3].
        v2f b = *(const v2f*)(xw + i * 4);   // ds_load_b64
        acc = __builtin_amdgcn_wmma_f32_16x16x4_f32(
                  /*neg_a=*/false, a, /*neg_b=*/false, b,
                  /*c_mod=*/(short)0, acc, /*reuse_a=*/false, /*reuse_b=*/false);
    }

    // Column N=0 of C: lane 0 holds M=v in acc[v] (v=0..7), lane 16 holds M=v+8.
    if (lane == 0) {
        #pragma unroll
        for (int v = 0; v < 8; ++v) red[wave * 16 + v] = acc[v];
    }
    if (lane == 16) {
        #pragma unroll
        for (int v = 0; v < 8; ++v) red[wave * 16 + 8 + v] = acc[v];
    }
    __syncthreads();

    if (threadIdx.x < 16) {
        float s = 0.0f;
        #pragma unroll
        for (int w = 0; w < WAVES; ++w) s += red[w * 16 + threadIdx.x];
        partials[(size_t)chunk * ROWS_PAD + tile * 16 + threadIdx.x] = s;
    }
}

// ---------------------------------------------------------------------------
// Stage 2a: y[r] = bias[r] + sum over 32 K-chunk partials.
// ---------------------------------------------------------------------------
__global__ void __launch_bounds__(256)
reduce_y(const float* __restrict__ partials, const float* __restrict__ bias,
         float* __restrict__ y) {
    const int r = blockIdx.x * blockDim.x + threadIdx.x;
    if (r >= ROWS_PAD) return;
    float s = 0.0f;
    #pragma unroll 4
    for (int c = 0; c < KCHUNKS; ++c) s += partials[(size_t)c * ROWS_PAD + r];
    y[r] = (r < NKERS) ? s + bias[r] : 0.0f;
}

// ---------------------------------------------------------------------------
// Stage 2b: z_partial[s][j] = sum_{i in slice s} (j < kwidths[i]) * y[i]*K[i][j]
// Coalesced row reads of `kernels`; 8 slices * 2 blocks = 16 WGs so the 1 MB
// kernel bank doesn't serialize behind stage 1.
// ---------------------------------------------------------------------------
__global__ void __launch_bounds__(256)
expand_partial(const float* __restrict__ y, const float* __restrict__ kern,
               const int* __restrict__ kw, float* __restrict__ zpart) {
    const int j = blockIdx.x * blockDim.x + threadIdx.x;
    const int s = blockIdx.y;
    if (j >= NKERS) return;
    const int i0 = s * ZSLICE_W;
    const int i1 = (i0 + ZSLICE_W < NKERS) ? i0 + ZSLICE_W : NKERS;
    float acc = 0.0f;
    for (int i = i0; i < i1; ++i) {
        const float yi = y[i];                    // scalar-broadcast load
        const int   w  = kw[i];                   // scalar-broadcast load
        const float kv = (j < w) ? kern[(size_t)i * NKERS + j] : 0.0f;
        acc = fmaf(yi, kv, acc);
    }
    zpart[(size_t)s * ROWS_PAD + j] = acc;
}

// ---------------------------------------------------------------------------
// Stage 2c: z[j] = sum over 8 slices.
// ---------------------------------------------------------------------------
__global__ void __launch_bounds__(256)
reduce_z(const float* __restrict__ zpart, float* __restrict__ z) {
    const int j = blockIdx.x * blockDim.x + threadIdx.x;
    if (j >= NKERS) return;
    float s = 0.0f;
    #pragma unroll
    for (int c = 0; c < ZSLICES; ++c) s += zpart[(size_t)c * ROWS_PAD + j];
    z[j] = s;
}

// ---------------------------------------------------------------------------
// d_in order (setup_inputs): 0=input[65536] f32, 1=weight[500*65536] f32,
// 2=bias[500] f32, 3=kernels[500*500] f32, 4=kwidths[500] i32.
// d_ws usage: partials 32*512 f32 | y 512 f32 | zpart 8*512 f32  (~84 KB).
// ---------------------------------------------------------------------------
extern "C" void kernel_launch(void* const* d_in, const int* in_sizes, int n_in,
                              void* d_out, int out_size, void* d_ws, size_t ws_size,
                              hipStream_t stream) {
    const float* x    = (const float*)d_in[0];
    const float* W    = (const float*)d_in[1];
    const float* bias = (const float*)d_in[2];
    const float* kern = (const float*)d_in[3];
    const int*   kw   = (const int*)d_in[4];
    float*       z    = (float*)d_out;

    float* partials = (float*)d_ws;                          // KCHUNKS * ROWS_PAD
    float* y        = partials + (size_t)KCHUNKS * ROWS_PAD; // ROWS_PAD
    float* zpart    = y + ROWS_PAD;                          // ZSLICES * ROWS_PAD

    dim3 g1(ROW_TILES, KCHUNKS);                             // 32 x 32 workgroups
    gemv_partials_wmma<<<g1, 256, 0, stream>>>(W, x, partials);

    reduce_y<<<(ROWS_PAD + 255) / 256, 256, 0, stream>>>(partials, bias, y);

    dim3 g3((NKERS + 255) / 256, ZSLICES);                   // 2 x 8 workgroups
    expand_partial<<<g3, 256, 0, stream>>>(y, kern, kw, zpart);

    reduce_z<<<(NKERS + 255) / 256, 256, 0, stream>>>(zpart, z);
}